// MF_VSS_19774029431787
// MI455X (gfx1250) — compile-verified
//
#include <hip/hip_runtime.h>
#include <hip/hip_bf16.h>
#include <math.h>

#define DEV __device__ __forceinline__

typedef float v2f __attribute__((ext_vector_type(2)));
typedef float v8f __attribute__((ext_vector_type(8)));

constexpr int B_  = 2;
constexpr int C_  = 128;
constexpr int H_  = 64;
constexpr int W_  = 64;
constexpr int L_  = H_ * W_;        // 4096
constexpr int HK  = H_ / 2;
constexpr int WK  = W_ / 2;
constexpr int LK  = HK * WK;        // 1024
constexpr int NH  = 8;              // heads
constexpr int HD  = 16;             // head dim
constexpr int DI  = 256;            // d_inner
constexpr int DSN = 16;             // d_state
constexpr int DTR = 8;              // dt_rank
constexpr int KD  = 4;              // directions
constexpr int XC  = DTR + 2 * DSN;  // 40

// ---------------- WMMA helpers (fp32 16x16x4) ----------------
DEV v8f wmma4(v2f a, v2f b, v8f c) {
  // D = A(16x4 f32) * B(4x16 f32) + C(16x16 f32)
  return __builtin_amdgcn_wmma_f32_16x16x4_f32(false, a, false, b, (short)0, c,
                                               false, false);
}

// A-fragment: lane (k/2)*16+m holds A[m][k0+2*(lane>>4)+v]
// B-fragment: lane (k/2)*16+n holds B[k][n] at same k mapping
DEV v8f gemm_loop(const float* Arow, int sAk, const float* Brow, int sBk,
                  int K, int kh2, v8f acc) {
  for (int k0 = 0; k0 < K; k0 += 4) {
    v2f a, b;
    a.x = Arow[(size_t)(k0 + kh2) * sAk];
    a.y = Arow[(size_t)(k0 + kh2 + 1) * sAk];
    b.x = Brow[(size_t)(k0 + kh2) * sBk];
    b.y = Brow[(size_t)(k0 + kh2 + 1) * sBk];
    acc = wmma4(a, b, acc);
  }
  return acc;
}

DEV float siluf(float x) { return x / (1.f + __expf(-x)); }
DEV float softplusf(float x) { return fmaxf(x, 0.f) + log1pf(__expf(-fabsf(x))); }

DEV int perm_pos(int dir, int t) {
  int tt = (dir >= 2) ? (L_ - 1 - t) : t;
  if (dir & 1) return ((tt & (H_ - 1)) << 6) | (tt >> 6);  // (t%H)*W + t/H
  return tt;
}

// ---------------- 1) q projection: q[b,l,d] = x[b,:,l].qw[d,:] + qb ---------
__global__ void k_qproj(const float* __restrict__ x, const float* __restrict__ qw,
                        const float* __restrict__ qb, float* __restrict__ q) {
  int lane = threadIdx.x & 31, mh = lane & 15, g = lane >> 4, kh2 = g * 2;
  int mt = blockIdx.x, nt = blockIdx.y, b = blockIdx.z;
  const float* Arow = x + (size_t)b * C_ * L_ + (mt * 16 + mh);  // +k*L_
  const float* Brow = qw + (size_t)(nt * 16 + mh) * C_;
  v8f acc = {0.f,0.f,0.f,0.f,0.f,0.f,0.f,0.f};
  acc = gemm_loop(Arow, L_, Brow, 1, C_, kh2, acc);
  float bias = qb[nt * 16 + mh];
  float* ob = q + ((size_t)b * L_ + mt * 16) * C_ + nt * 16 + mh;
#pragma unroll
  for (int i = 0; i < 8; ++i) ob[(size_t)(i + 8 * g) * C_] = acc[i] + bias;
}

// ---------------- 2) im2col for 2x2 stride-2 conv ----------------
__global__ void k_im2col(const float* __restrict__ sc, float* __restrict__ col) {
  int idx = blockIdx.x * blockDim.x + threadIdx.x;
  const int KTOT = C_ * 4;
  int total = B_ * LK * KTOT;
  if (idx >= total) return;
  int k = idx % KTOT, rest = idx / KTOT;
  int lk = rest % LK, b = rest / LK;
  int ci = k >> 2, dh = (k >> 1) & 1, dw = k & 1;
  int hk = lk / WK, wk = lk % WK;
  col[idx] = sc[(((size_t)b * C_ + ci) * H_ + (hk * 2 + dh)) * W_ + (wk * 2 + dw)];
}

// ---------------- 3) SR conv GEMM: scr = col @ srw^T + srb ----------------
__global__ void k_gemm_sr(const float* __restrict__ col, const float* __restrict__ srw,
                          const float* __restrict__ srb, float* __restrict__ scr) {
  int lane = threadIdx.x & 31, mh = lane & 15, g = lane >> 4, kh2 = g * 2;
  int mt = blockIdx.x, nt = blockIdx.y, b = blockIdx.z;
  const float* Arow = col + ((size_t)b * LK + mt * 16 + mh) * 512;
  const float* Brow = srw + (size_t)(nt * 16 + mh) * 512;
  v8f acc = {0.f,0.f,0.f,0.f,0.f,0.f,0.f,0.f};
  acc = gemm_loop(Arow, 1, Brow, 1, 512, kh2, acc);
  float bias = srb[nt * 16 + mh];
  float* ob = scr + ((size_t)b * LK + mt * 16) * C_ + nt * 16 + mh;
#pragma unroll
  for (int i = 0; i < 8; ++i) ob[(size_t)(i + 8 * g) * C_] = acc[i] + bias;
}

// ---------------- 4) row LayerNorm in-place ----------------
__global__ void k_ln_rows(float* __restrict__ buf, const float* __restrict__ g,
                          const float* __restrict__ bb, int rows, int dim) {
  int r = blockIdx.x * blockDim.x + threadIdx.x;
  if (r >= rows) return;
  float* p = buf + (size_t)r * dim;
  float mu = 0.f;
  for (int i = 0; i < dim; ++i) mu += p[i];
  mu /= dim;
  float var = 0.f;
  for (int i = 0; i < dim; ++i) { float d = p[i] - mu; var += d * d; }
  var /= dim;
  float rs = rsqrtf(var + 1e-5f);
  for (int i = 0; i < dim; ++i) p[i] = (p[i] - mu) * rs * g[i] + bb[i];
}

// ---------------- 5) kv GEMM: kvb = scr @ kvw^T + kvb ----------------
__global__ void k_gemm_kv(const float* __restrict__ scr, const float* __restrict__ kvw,
                          const float* __restrict__ kvbias, float* __restrict__ kvb) {
  int lane = threadIdx.x & 31, mh = lane & 15, g = lane >> 4, kh2 = g * 2;
  int mt = blockIdx.x, nt = blockIdx.y, b = blockIdx.z;
  const float* Arow = scr + ((size_t)b * LK + mt * 16 + mh) * C_;
  const float* Brow = kvw + (size_t)(nt * 16 + mh) * C_;
  v8f acc = {0.f,0.f,0.f,0.f,0.f,0.f,0.f,0.f};
  acc = gemm_loop(Arow, 1, Brow, 1, C_, kh2, acc);
  float bias = kvbias[nt * 16 + mh];
  float* ob = kvb + ((size_t)b * LK + mt * 16) * (2 * C_) + nt * 16 + mh;
#pragma unroll
  for (int i = 0; i < 8; ++i) ob[(size_t)(i + 8 * g) * (2 * C_)] = acc[i] + bias;
}

// ---------------- 6) flash attention (per wave: 16 q rows, one head) --------
__global__ void k_attn(const float* __restrict__ q, const float* __restrict__ kv,
                       float* __restrict__ o) {
  int qt = blockIdx.x, hh = blockIdx.y, b = blockIdx.z;
  int lane = threadIdx.x & 31, mh = lane & 15, g = lane >> 4, kh2 = g * 2;
  __shared__ float P[16][17];

  v2f qa[4];
  const float* qbase = q + ((size_t)b * L_ + qt * 16 + mh) * C_ + hh * HD;
#pragma unroll
  for (int s = 0; s < 4; ++s) { qa[s].x = qbase[4 * s + kh2]; qa[s].y = qbase[4 * s + kh2 + 1]; }

  float m_i[8], l_i[8];
  v8f oacc = {0.f,0.f,0.f,0.f,0.f,0.f,0.f,0.f};
#pragma unroll
  for (int i = 0; i < 8; ++i) { m_i[i] = -1e30f; l_i[i] = 0.f; }

  for (int kt = 0; kt < LK / 16; ++kt) {
    // S = q @ k^T (K = head dim 16)
    v8f S = {0.f,0.f,0.f,0.f,0.f,0.f,0.f,0.f};
    const float* kbase = kv + ((size_t)b * LK + kt * 16 + mh) * (2 * C_) + hh * HD;
#pragma unroll
    for (int s = 0; s < 4; ++s) {
      v2f bf; bf.x = kbase[4 * s + kh2]; bf.y = kbase[4 * s + kh2 + 1];
      S = wmma4(qa[s], bf, S);
    }
    // online softmax (row stats reduced across the 16-lane half-groups)
#pragma unroll
    for (int i = 0; i < 8; ++i) {
      float sv = S[i] * 0.25f;  // KEY_DIM^-0.5
      float rm = sv;
      for (int off = 1; off < 16; off <<= 1) rm = fmaxf(rm, __shfl_xor(rm, off, 32));
      float mn = fmaxf(m_i[i], rm);
      float p = __expf(sv - mn);
      float rs = p;
      for (int off = 1; off < 16; off <<= 1) rs += __shfl_xor(rs, off, 32);
      float corr = __expf(m_i[i] - mn);
      l_i[i] = l_i[i] * corr + rs;
      m_i[i] = mn;
      oacc[i] *= corr;
      S[i] = p;
    }
    // relay P (C layout) -> A layout through LDS
    __syncthreads();
#pragma unroll
    for (int i = 0; i < 8; ++i) P[i + 8 * g][mh] = S[i];
    __syncthreads();
    // oacc += P @ V (K = 16 keys)
    const float* vbase = kv + ((size_t)b * LK + kt * 16) * (2 * C_) + C_ + hh * HD + mh;
#pragma unroll
    for (int s = 0; s < 4; ++s) {
      v2f a, bf;
      a.x = P[mh][4 * s + kh2];
      a.y = P[mh][4 * s + kh2 + 1];
      bf.x = vbase[(size_t)(4 * s + kh2) * (2 * C_)];
      bf.y = vbase[(size_t)(4 * s + kh2 + 1) * (2 * C_)];
      oacc = wmma4(a, bf, oacc);
    }
  }
  float* ob = o + ((size_t)b * L_ + qt * 16) * C_ + hh * HD + mh;
#pragma unroll
  for (int i = 0; i < 8; ++i) ob[(size_t)(i + 8 * g) * C_] = oacc[i] / l_i[i];
}

// ---------------- 7) proj + residual -> x1 (B,L,C NHWC) ----------------
__global__ void k_proj_res(const float* __restrict__ o, const float* __restrict__ pw,
                           const float* __restrict__ pb, const float* __restrict__ xin,
                           float* __restrict__ x1) {
  int lane = threadIdx.x & 31, mh = lane & 15, g = lane >> 4, kh2 = g * 2;
  int mt = blockIdx.x, nt = blockIdx.y, b = blockIdx.z;
  const float* Arow = o + ((size_t)b * L_ + mt * 16 + mh) * C_;
  const float* Brow = pw + (size_t)(nt * 16 + mh) * C_;
  v8f acc = {0.f,0.f,0.f,0.f,0.f,0.f,0.f,0.f};
  acc = gemm_loop(Arow, 1, Brow, 1, C_, kh2, acc);
  int c = nt * 16 + mh;
  float bias = pb[c];
#pragma unroll
  for (int i = 0; i < 8; ++i) {
    int l = mt * 16 + i + 8 * g;
    x1[((size_t)b * L_ + l) * C_ + c] =
        acc[i] + bias + xin[((size_t)b * C_ + c) * L_ + l];
  }
}

// ---------------- 8) in_proj GEMM: split into xx / z ----------------
__global__ void k_inproj(const float* __restrict__ x1, const float* __restrict__ iw,
                         const float* __restrict__ ib, float* __restrict__ xx,
                         float* __restrict__ z) {
  int lane = threadIdx.x & 31, mh = lane & 15, g = lane >> 4, kh2 = g * 2;
  int mt = blockIdx.x, nt = blockIdx.y, b = blockIdx.z;
  const float* Arow = x1 + ((size_t)b * L_ + mt * 16 + mh) * C_;
  const float* Brow = iw + (size_t)(nt * 16 + mh) * C_;
  v8f acc = {0.f,0.f,0.f,0.f,0.f,0.f,0.f,0.f};
  acc = gemm_loop(Arow, 1, Brow, 1, C_, kh2, acc);
  int ng = nt * 16 + mh;
  float bias = ib[ng];
  float* dst = (ng < DI) ? (xx + ((size_t)b * L_ + mt * 16) * DI + ng)
                         : (z + ((size_t)b * L_ + mt * 16) * DI + (ng - DI));
#pragma unroll
  for (int i = 0; i < 8; ++i) dst[(size_t)(i + 8 * g) * DI] = acc[i] + bias;
}

// ---------------- 9) depthwise 3x3 SAME + SiLU ----------------
__global__ void k_dwconv(const float* __restrict__ xx, const float* __restrict__ cw,
                         const float* __restrict__ cb, float* __restrict__ xc) {
  int bl = blockIdx.x;
  int d = threadIdx.x;
  int b = bl / L_, l = bl % L_;
  int h = l / W_, w = l % W_;
  float s = cb[d];
#pragma unroll
  for (int dh = -1; dh <= 1; ++dh)
#pragma unroll
    for (int dw = -1; dw <= 1; ++dw) {
      int hh = h + dh, ww = w + dw;
      if (hh < 0 || hh >= H_ || ww < 0 || ww >= W_) continue;
      s += xx[((size_t)b * L_ + hh * W_ + ww) * DI + d] * cw[d * 9 + (dh + 1) * 3 + (dw + 1)];
    }
  xc[((size_t)b * L_ + l) * DI + d] = siluf(s);
}

// ---------------- 10) x_dbl GEMM per direction (N=40, K=256) ----------------
__global__ void k_xdbl(const float* __restrict__ xc, const float* __restrict__ xpw,
                       float* __restrict__ xdbl) {
  int lane = threadIdx.x & 31, mh = lane & 15, g = lane >> 4, kh2 = g * 2;
  int mt = blockIdx.x, nt = blockIdx.y;
  int bk = blockIdx.z, b = bk >> 2, kdir = bk & 3;
  int t = mt * 16 + mh;
  int pos = perm_pos(kdir, t);
  const float* Arow = xc + ((size_t)b * L_ + pos) * DI;
  int nclamp = min(nt * 16 + mh, XC - 1);
  const float* Brow = xpw + ((size_t)kdir * XC + nclamp) * DI;
  v8f acc = {0.f,0.f,0.f,0.f,0.f,0.f,0.f,0.f};
  acc = gemm_loop(Arow, 1, Brow, 1, DI, kh2, acc);
  int c = nt * 16 + mh;
  if (c < XC) {
    float* ob = xdbl + ((size_t)(b * KD + kdir) * L_ + mt * 16) * XC + c;
#pragma unroll
    for (int i = 0; i < 8; ++i) ob[(size_t)(i + 8 * g) * XC] = acc[i];
  }
}

// ---------------- 11) selective scan (one block per (b,dir), lane=d) --------
__global__ void k_scan(const float* __restrict__ xdbl, const float* __restrict__ xc,
                       const float* __restrict__ dtw, const float* __restrict__ dtb,
                       const float* __restrict__ Alogs, const float* __restrict__ Dsp,
                       float* __restrict__ ybuf) {
  int bk = blockIdx.x, b = bk >> 2, kdir = bk & 3;
  int d = threadIdx.x;  // 0..255
  __shared__ float row[XC];
  float wr[DTR];
#pragma unroll
  for (int r = 0; r < DTR; ++r) wr[r] = dtw[((size_t)kdir * DI + d) * DTR + r];
  float bias = dtb[kdir * DI + d];
  float Ar[DSN];
#pragma unroll
  for (int n = 0; n < DSN; ++n)
    Ar[n] = -__expf(Alogs[((size_t)kdir * DI + d) * DSN + n]);
  float Dd = Dsp[kdir * DI + d];
  float h[DSN];
#pragma unroll
  for (int n = 0; n < DSN; ++n) h[n] = 0.f;

  const float* xdb = xdbl + (size_t)(b * KD + kdir) * L_ * XC;
  const float* xcb = xc + (size_t)b * L_ * DI;
  float* yb = ybuf + (size_t)(b * KD + kdir) * L_ * DI;

  for (int t = 0; t < L_; ++t) {
    if (threadIdx.x < XC) row[threadIdx.x] = xdb[(size_t)t * XC + threadIdx.x];
    __syncthreads();
    if (t + 1 < L_) __builtin_prefetch(xdb + (size_t)(t + 1) * XC, 0, 0);
    int pos = perm_pos(kdir, t);
    float u = xcb[(size_t)pos * DI + d];
    float acc = bias;
#pragma unroll
    for (int r = 0; r < DTR; ++r) acc += wr[r] * row[r];
    float dt = softplusf(acc);
    float dtu = dt * u;
    float y = 0.f;
#pragma unroll
    for (int n = 0; n < DSN; ++n) {
      h[n] = h[n] * __expf(dt * Ar[n]) + dtu * row[DTR + n];
      y += h[n] * row[DTR + DSN + n];
    }
    // cross-merge target position == read position (for all 4 directions)
    yb[(size_t)pos * DI + d] = y + Dd * u;
    __syncthreads();
  }
}

// ---------------- 12) merge 4 dirs + LN + SiLU gate ----------------
__global__ void k_merge_ln_gate(const float* __restrict__ ybuf, const float* __restrict__ z,
                                const float* __restrict__ g, const float* __restrict__ bb,
                                float* __restrict__ yg) {
  int bl = blockIdx.x;
  int b = bl / L_, l = bl % L_;
  int d = threadIdx.x;
  float s = 0.f;
#pragma unroll
  for (int k = 0; k < KD; ++k)
    s += ybuf[((size_t)(b * KD + k) * L_ + l) * DI + d];
  __shared__ float red[DI];
  red[d] = s;
  __syncthreads();
  for (int off = DI / 2; off > 0; off >>= 1) {
    if (d < off) red[d] += red[d + off];
    __syncthreads();
  }
  float mu = red[0] / DI;
  __syncthreads();
  float dv = s - mu;
  red[d] = dv * dv;
  __syncthreads();
  for (int off = DI / 2; off > 0; off >>= 1) {
    if (d < off) red[d] += red[d + off];
    __syncthreads();
  }
  float var = red[0] / DI;
  float val = dv * rsqrtf(var + 1e-5f) * g[d] + bb[d];
  float zz = z[((size_t)b * L_ + l) * DI + d];
  yg[((size_t)b * L_ + l) * DI + d] = val * siluf(zz);
}

// ---------------- 13) out_proj GEMM -> NCHW d_out ----------------
__global__ void k_outproj(const float* __restrict__ yg, const float* __restrict__ ow,
                          const float* __restrict__ ob, float* __restrict__ out) {
  int lane = threadIdx.x & 31, mh = lane & 15, g = lane >> 4, kh2 = g * 2;
  int mt = blockIdx.x, nt = blockIdx.y, b = blockIdx.z;
  const float* Arow = yg + ((size_t)b * L_ + mt * 16 + mh) * DI;
  const float* Brow = ow + (size_t)(nt * 16 + mh) * DI;
  v8f acc = {0.f,0.f,0.f,0.f,0.f,0.f,0.f,0.f};
  acc = gemm_loop(Arow, 1, Brow, 1, DI, kh2, acc);
  int c = nt * 16 + mh;
  float bias = ob[c];
#pragma unroll
  for (int i = 0; i < 8; ++i) {
    int l = mt * 16 + i + 8 * g;
    out[((size_t)b * C_ + c) * L_ + l] = acc[i] + bias;
  }
}

extern "C" void kernel_launch(void* const* d_in, const int* in_sizes, int n_in,
                              void* d_out, int out_size, void* d_ws, size_t ws_size,
                              hipStream_t stream) {
  const float* x_in    = (const float*)d_in[0];
  const float* sc_in   = (const float*)d_in[1];
  const float* q_w     = (const float*)d_in[2];
  const float* q_b     = (const float*)d_in[3];
  const float* sr_w    = (const float*)d_in[4];
  const float* sr_b    = (const float*)d_in[5];
  const float* sr_ln_g = (const float*)d_in[6];
  const float* sr_ln_b = (const float*)d_in[7];
  const float* kv_w    = (const float*)d_in[8];
  const float* kv_b    = (const float*)d_in[9];
  const float* proj_w  = (const float*)d_in[10];
  const float* proj_b  = (const float*)d_in[11];
  const float* in_w    = (const float*)d_in[12];
  const float* in_b    = (const float*)d_in[13];
  const float* conv_w  = (const float*)d_in[14];
  const float* conv_b  = (const float*)d_in[15];
  const float* xproj_w = (const float*)d_in[16];
  const float* dtp_w   = (const float*)d_in[17];
  const float* dtp_b   = (const float*)d_in[18];
  const float* A_logs  = (const float*)d_in[19];
  const float* Ds      = (const float*)d_in[20];
  const float* oln_g   = (const float*)d_in[21];
  const float* oln_b   = (const float*)d_in[22];
  const float* out_w   = (const float*)d_in[23];
  const float* out_b   = (const float*)d_in[24];

  float* ws = (float*)d_ws;
  size_t off = 0;
  auto alloc = [&](size_t n) { float* p = ws + off; off += n; return p; };
  float* qb   = alloc((size_t)B_ * L_ * C_);       // q          4 MB
  float* col  = alloc((size_t)B_ * LK * 512);      // im2col     4 MB
  float* scr  = alloc((size_t)B_ * LK * C_);       // sr out     1 MB
  float* kvb  = alloc((size_t)B_ * LK * 2 * C_);   // k|v        2 MB
  float* ob   = alloc((size_t)B_ * L_ * C_);       // attn out   4 MB
  float* x1   = alloc((size_t)B_ * L_ * C_);       // mcfa out   4 MB
  float* xxb  = alloc((size_t)B_ * L_ * DI);       // xx         8 MB
  float* zb   = alloc((size_t)B_ * L_ * DI);       // z          8 MB
  float* xcb  = alloc((size_t)B_ * L_ * DI);       // conv+silu  8 MB
  float* xdb  = alloc((size_t)B_ * KD * L_ * XC);  // x_dbl      5 MB
  float* ybuf = alloc((size_t)B_ * KD * L_ * DI);  // per-dir y 32 MB
  float* ygb  = alloc((size_t)B_ * L_ * DI);       // gated y    8 MB
  (void)ws_size; (void)in_sizes; (void)n_in; (void)out_size;

  // ---- MCFA ----
  k_qproj<<<dim3(L_ / 16, C_ / 16, B_), 32, 0, stream>>>(x_in, q_w, q_b, qb);
  k_im2col<<<(B_ * LK * 512 + 255) / 256, 256, 0, stream>>>(sc_in, col);
  k_gemm_sr<<<dim3(LK / 16, C_ / 16, B_), 32, 0, stream>>>(col, sr_w, sr_b, scr);
  k_ln_rows<<<(B_ * LK + 255) / 256, 256, 0, stream>>>(scr, sr_ln_g, sr_ln_b, B_ * LK, C_);
  k_gemm_kv<<<dim3(LK / 16, (2 * C_) / 16, B_), 32, 0, stream>>>(scr, kv_w, kv_b, kvb);
  k_attn<<<dim3(L_ / 16, NH, B_), 32, 0, stream>>>(qb, kvb, ob);
  k_proj_res<<<dim3(L_ / 16, C_ / 16, B_), 32, 0, stream>>>(ob, proj_w, proj_b, x_in, x1);

  // ---- SS2D ----
  k_inproj<<<dim3(L_ / 16, (2 * DI) / 16, B_), 32, 0, stream>>>(x1, in_w, in_b, xxb, zb);
  k_dwconv<<<B_ * L_, DI, 0, stream>>>(xxb, conv_w, conv_b, xcb);
  k_xdbl<<<dim3(L_ / 16, 3, B_ * KD), 32, 0, stream>>>(xcb, xproj_w, xdb);
  k_scan<<<B_ * KD, DI, 0, stream>>>(xdb, xcb, dtp_w, dtp_b, A_logs, Ds, ybuf);
  k_merge_ln_gate<<<B_ * L_, DI, 0, stream>>>(ybuf, zb, oln_g, oln_b, ygb);
  k_outproj<<<dim3(L_ / 16, C_ / 16, B_), 32, 0, stream>>>(ygb, out_w, out_b, (float*)d_out);
}